// RetraceLossRecursive_83382495085148
// MI455X (gfx1250) — compile-verified
//
#include <hip/hip_runtime.h>

// ---------------------------------------------------------------------------
// RetraceLossRecursive for MI455X (gfx1250, wave32)
//
// Segmented backward linear-recurrence scan, with the CDNA5 Tensor Data Mover
// (TENSOR_LOAD_TO_LDS, TENSORcnt) staging each chunk as a 2D tile into LDS,
// double-buffered. TDM hardware padding (8 dwords per 64) gives the
// bank-conflict-free row stride of 72 dwords.
//
//   ret[t] = r'[t] + a[t] * ret[t+1]
//   a[t]  = G * c[t+1]                  (t < NT-1),   a[NT-1] = 0
//   r'[t] = r[t] + G*V[t+1] - a*Q[t+1]  (t < NT-1)
//   r'[NT-1] = r + G*(V + c*Q)
//   c[t]  = exp(min(tlogp[t] - ologp[t], 0))
// then huber-mean of (q[:, :-1] - ret).
// ---------------------------------------------------------------------------

static constexpr int   B_     = 2048;
static constexpr int   T_     = 2049;
static constexpr int   K_     = 8;
static constexpr int   NT     = T_ - 1;           // 2048 scan steps
static constexpr int   NC     = B_ * K_;          // 16384 chains
static constexpr int   NSEG   = 32;
static constexpr int   SEGLEN = NT / NSEG;        // 64
static constexpr int   BLK    = 256;
static constexpr int   P13_BLOCKS = (NC * NSEG) / BLK;   // 2048
static constexpr int   CH     = 8;                // time steps per staged chunk
static constexpr int   NCHUNK = SEGLEN / CH;      // 8
static constexpr int   ROWDW  = CH * K_ + 8;      // 72 dwords: 64 data + 8 TDM pad
static constexpr int   TILEDW = 32 * ROWDW;       // 2304 dwords per stream tile
static constexpr int   OLPDW  = 32 * CH;          // 256 dwords
static constexpr int   BUFDW1 = 4 * TILEDW + OLPDW;   // 9472  (pass1 buffer, dwords)
static constexpr int   BUFDW3 = 5 * TILEDW + OLPDW;   // 11776 (pass3 buffer, dwords)
static constexpr float GAMMA  = 0.99f;

typedef unsigned int v4u __attribute__((ext_vector_type(4)));
typedef unsigned int v8u __attribute__((ext_vector_type(8)));

// ---- Tensor DMA descriptor group1 (256b), 2D tile --------------------------
// w0: workgroup_mask=0 | data_size(17:16) | pad_enable(20) | pad_interval(24:22)
//     | pad_amount(31:25)
// pad_interval=5 -> pad every 64 dwords; pad_amount=7 -> 8 dwords of pad.
static constexpr unsigned G1W0_PAD   = (2u << 16) | (1u << 20) | (5u << 22) | (7u << 25);
static constexpr unsigned G1W0_NOPAD = (2u << 16);

__device__ __forceinline__ v8u make_g1(unsigned w0, unsigned td0, unsigned td1,
                                       unsigned tile0, unsigned tile1, unsigned stride0)
{
    v8u g;
    g[0] = w0;
    g[1] = td0 << 16;      // [47:32] atomic_barrier_addr=0 | [63:48] tensor_dim0[15:0]
    g[2] = td1 << 16;      // [79:64] tensor_dim0[31:16]=0  | [95:80] tensor_dim1[15:0]
    g[3] = tile0 << 16;    // [111:96] tensor_dim1[31:16]=0 | [127:112] tile_dim0
    g[4] = tile1;          // [143:128] tile_dim1 | [159:144] tile_dim2=0 (2D)
    g[5] = stride0;        // [191:160] tensor_dim0_stride[31:0]
    g[6] = 0;              // stride0[47:32]=0 | tensor_dim1_stride[15:0]=0
    g[7] = 0;
    return g;
}

// One TDM 2D tile load: group0 built from (lds_addr, global byte address).
__device__ __forceinline__ void tdm_load_2d(unsigned ldsAddr, const void* gptr, v8u g1)
{
    const unsigned long long ga = (unsigned long long)gptr;
    v4u g0;
    g0[0] = 1u;                                     // count=1 (valid), no gather
    g0[1] = ldsAddr;                                // lds_addr (bytes)
    g0[2] = (unsigned)ga;                           // global_addr[31:0]
    g0[3] = (unsigned)((ga >> 32) & 0x01FFFFFFu)    // global_addr[56:32]
          | (2u << 30);                             // type=2 ("image")
    asm volatile("tensor_load_to_lds %0, %1" :: "s"(g0), "s"(g1) : "memory");
}

#define WAIT_TENSORCNT(n) __builtin_amdgcn_s_wait_tensorcnt(n)

// ---------------------------------------------------------------------------
// Pass 1: per (chain, segment) compose the recurrence into (A, Bv):
//         ret[segStart] = Bv + A * ret[segEnd]
// ---------------------------------------------------------------------------
__global__ __launch_bounds__(BLK) void retrace_pass1(
    const float* __restrict__ tq, const float* __restrict__ tv,
    const float* __restrict__ rw, const float* __restrict__ olp,
    const float* __restrict__ tlp,
    float* __restrict__ Aout, float* __restrict__ Bout)
{
    extern __shared__ float smem[];
    const unsigned smemBase = (unsigned)(size_t)smem;   // low 32 bits = LDS byte offset
    const int tid    = threadIdx.x;
    const int s      = blockIdx.x >> 6;                 // segment (64 blocks per segment)
    const int bGlob0 = (blockIdx.x & 63) * 32;
    const int t0     = s * SEGLEN;
    const int b_loc  = tid >> 3, k = tid & 7;
    const int rowOff = b_loc * ROWDW + k;

    const v8u g1Big = make_g1(G1W0_PAD,   CH * K_, 32, CH * K_, 32, (unsigned)(T_ * K_));
    const v8u g1Olp = make_g1(G1W0_NOPAD, CH,      32, CH,      32, (unsigned)T_);

    // chunk j covers t in [lo, lo+CH), processed right->left; issued by wave 0 only
    auto issue = [&](int j) {
        const int lo = t0 + SEGLEN - (j + 1) * CH;
        const unsigned base = smemBase + (unsigned)((j & 1) * BUFDW1 * 4);
        const size_t e0 = ((size_t)bGlob0 * T_ + lo) * K_;      // rewards tile (t)
        const size_t e1 = e0 + K_;                              // t+1 tiles
        tdm_load_2d(base + 0u * TILEDW * 4u, rw  + e0, g1Big);
        tdm_load_2d(base + 1u * TILEDW * 4u, tv  + e1, g1Big);
        tdm_load_2d(base + 2u * TILEDW * 4u, tq  + e1, g1Big);
        tdm_load_2d(base + 3u * TILEDW * 4u, tlp + e1, g1Big);
        tdm_load_2d(base + 4u * TILEDW * 4u, olp + (size_t)bGlob0 * T_ + lo + 1, g1Olp);
    };

    if (tid < 32) issue(0);

    float A = 1.0f, Bv = 0.0f;
    for (int j = 0; j < NCHUNK; ++j) {
        if (tid < 32) {
            if (j + 1 < NCHUNK) { issue(j + 1); WAIT_TENSORCNT(5); }
            else                { WAIT_TENSORCNT(0); }
        }
        __syncthreads();
        const float* buf = smem + (j & 1) * BUFDW1;
        const bool lastChunk = (s == NSEG - 1) && (j == 0);
#pragma unroll
        for (int i = 0; i < CH; ++i) {
            const int t_loc = CH - 1 - i;
            const float r  = buf[0 * TILEDW + rowOff + t_loc * K_];
            const float V  = buf[1 * TILEDW + rowOff + t_loc * K_];
            const float Qt = buf[2 * TILEDW + rowOff + t_loc * K_];
            const float tl = buf[3 * TILEDW + rowOff + t_loc * K_];
            const float ol = buf[4 * TILEDW + b_loc * CH + t_loc];
            const float cc = __expf(fminf(tl - ol, 0.0f));
            float a, rp;
            if (lastChunk && i == 0) { a = 0.0f; rp = fmaf(GAMMA, fmaf(cc, Qt, V), r); }
            else { a = GAMMA * cc; rp = fmaf(GAMMA, V, r) - a * Qt; }
            Bv = fmaf(a, Bv, rp);
            A *= a;
        }
        __syncthreads();
    }
    const int g = blockIdx.x * BLK + tid;   // == s*NC + (b*8 + k)
    Aout[g] = A;
    Bout[g] = Bv;
}

// ---------------------------------------------------------------------------
// Pass 2: per chain, scan the NSEG (A,B) pairs right-to-left.
// edges[s*NC + c] = ret at the RIGHT boundary of segment s (its carry-in).
// ---------------------------------------------------------------------------
__global__ __launch_bounds__(BLK) void retrace_pass2(
    const float* __restrict__ Ain, const float* __restrict__ Bin,
    float* __restrict__ edges)
{
    const int c = blockIdx.x * BLK + threadIdx.x;
    float e = 0.0f;
#pragma unroll
    for (int s = NSEG - 1; s >= 0; --s) {
        edges[s * NC + c] = e;
        e = fmaf(Ain[s * NC + c], e, Bin[s * NC + c]);
    }
}

// ---------------------------------------------------------------------------
// Pass 3: recompute within-segment ret from the carry, accumulate Huber loss,
// block-reduce to one partial per block.
// ---------------------------------------------------------------------------
__global__ __launch_bounds__(BLK) void retrace_pass3(
    const float* __restrict__ q,
    const float* __restrict__ tq, const float* __restrict__ tv,
    const float* __restrict__ rw, const float* __restrict__ olp,
    const float* __restrict__ tlp,
    const float* __restrict__ edges,
    float* __restrict__ partials)
{
    extern __shared__ float smem[];
    const unsigned smemBase = (unsigned)(size_t)smem;
    const int tid    = threadIdx.x;
    const int s      = blockIdx.x >> 6;
    const int bGlob0 = (blockIdx.x & 63) * 32;
    const int t0     = s * SEGLEN;
    const int b_loc  = tid >> 3, k = tid & 7;
    const int rowOff = b_loc * ROWDW + k;
    const int g      = blockIdx.x * BLK + tid;

    const v8u g1Big = make_g1(G1W0_PAD,   CH * K_, 32, CH * K_, 32, (unsigned)(T_ * K_));
    const v8u g1Olp = make_g1(G1W0_NOPAD, CH,      32, CH,      32, (unsigned)T_);

    auto issue = [&](int j) {
        const int lo = t0 + SEGLEN - (j + 1) * CH;
        const unsigned base = smemBase + (unsigned)((j & 1) * BUFDW3 * 4);
        const size_t e0 = ((size_t)bGlob0 * T_ + lo) * K_;
        const size_t e1 = e0 + K_;
        tdm_load_2d(base + 0u * TILEDW * 4u, rw  + e0, g1Big);
        tdm_load_2d(base + 1u * TILEDW * 4u, tv  + e1, g1Big);
        tdm_load_2d(base + 2u * TILEDW * 4u, tq  + e1, g1Big);
        tdm_load_2d(base + 3u * TILEDW * 4u, tlp + e1, g1Big);
        tdm_load_2d(base + 4u * TILEDW * 4u, q   + e0, g1Big);
        tdm_load_2d(base + 5u * TILEDW * 4u, olp + (size_t)bGlob0 * T_ + lo + 1, g1Olp);
    };

    if (tid < 32) issue(0);

    float ret  = edges[g];
    float loss = 0.0f;
    for (int j = 0; j < NCHUNK; ++j) {
        if (tid < 32) {
            if (j + 1 < NCHUNK) { issue(j + 1); WAIT_TENSORCNT(6); }
            else                { WAIT_TENSORCNT(0); }
        }
        __syncthreads();
        const float* buf = smem + (j & 1) * BUFDW3;
        const bool lastChunk = (s == NSEG - 1) && (j == 0);
#pragma unroll
        for (int i = 0; i < CH; ++i) {
            const int t_loc = CH - 1 - i;
            const float r  = buf[0 * TILEDW + rowOff + t_loc * K_];
            const float V  = buf[1 * TILEDW + rowOff + t_loc * K_];
            const float Qt = buf[2 * TILEDW + rowOff + t_loc * K_];
            const float tl = buf[3 * TILEDW + rowOff + t_loc * K_];
            const float qv = buf[4 * TILEDW + rowOff + t_loc * K_];
            const float ol = buf[5 * TILEDW + b_loc * CH + t_loc];
            const float cc = __expf(fminf(tl - ol, 0.0f));
            float a, rp;
            if (lastChunk && i == 0) { a = 0.0f; rp = fmaf(GAMMA, fmaf(cc, Qt, V), r); }
            else { a = GAMMA * cc; rp = fmaf(GAMMA, V, r) - a * Qt; }
            ret = fmaf(a, ret, rp);
            const float d  = qv - ret;
            const float ad = fabsf(d);
            loss += (ad < 1.0f) ? (0.5f * d * d) : (ad - 0.5f);
        }
        __syncthreads();
    }

    __shared__ float red[BLK];
    red[tid] = loss;
    __syncthreads();
#pragma unroll
    for (int off = BLK / 2; off > 0; off >>= 1) {
        if (tid < off) red[tid] += red[tid + off];
        __syncthreads();
    }
    if (tid == 0) partials[blockIdx.x] = red[0];
}

// ---------------------------------------------------------------------------
// Pass 4: deterministic final reduction (double accumulation) -> mean.
// ---------------------------------------------------------------------------
__global__ __launch_bounds__(BLK) void retrace_reduce(
    const float* __restrict__ partials, int n, float* __restrict__ out)
{
    __shared__ double sm[BLK];
    double acc = 0.0;
    for (int i = threadIdx.x; i < n; i += BLK) acc += (double)partials[i];
    sm[threadIdx.x] = acc;
    __syncthreads();
#pragma unroll
    for (int off = BLK / 2; off > 0; off >>= 1) {
        if (threadIdx.x < off) sm[threadIdx.x] += sm[threadIdx.x + off];
        __syncthreads();
    }
    if (threadIdx.x == 0) {
        const double denom = (double)B_ * (double)NT * (double)K_;
        out[0] = (float)(sm[0] / denom);
    }
}

// ---------------------------------------------------------------------------
extern "C" void kernel_launch(void* const* d_in, const int* in_sizes, int n_in,
                              void* d_out, int out_size, void* d_ws, size_t ws_size,
                              hipStream_t stream)
{
    (void)in_sizes; (void)n_in; (void)out_size; (void)ws_size;
    const float* q   = (const float*)d_in[0];  // state_trajectory_action_values          (B,T,K)
    const float* tq  = (const float*)d_in[1];  // target_state_trajectory_action_values   (B,T,K)
    const float* tv  = (const float*)d_in[2];  // target_expected_state_values            (B,T,K)
    const float* rw  = (const float*)d_in[3];  // rewards                                 (B,T,K)
    const float* olp = (const float*)d_in[4];  // original_log_trajectory_action_probs    (B,T)
    const float* tlp = (const float*)d_in[5];  // target_log_trajectory_task_action_probs (B,T,K)
    float* out = (float*)d_out;

    // Workspace layout (floats): A | B | edges | partials  (~6.3 MB total)
    float* wsf      = (float*)d_ws;
    float* A        = wsf;
    float* Bv       = A     + NC * NSEG;
    float* edges    = Bv    + NC * NSEG;
    float* partials = edges + NC * NSEG;

    const size_t shm1 = (size_t)(2 * BUFDW1) * 4;   // 75776 B
    const size_t shm3 = (size_t)(2 * BUFDW3) * 4;   // 94208 B

    retrace_pass1<<<P13_BLOCKS, BLK, shm1, stream>>>(tq, tv, rw, olp, tlp, A, Bv);
    retrace_pass2<<<NC / BLK, BLK, 0, stream>>>(A, Bv, edges);
    retrace_pass3<<<P13_BLOCKS, BLK, shm3, stream>>>(q, tq, tv, rw, olp, tlp, edges, partials);
    retrace_reduce<<<1, BLK, 0, stream>>>(partials, P13_BLOCKS, out);
}